// Biaffine_29661044146244
// MI455X (gfx1250) — compile-verified
//
#include <hip/hip_runtime.h>

// Biaffine fused kernel for MI455X (gfx1250, wave32, WMMA).
//
//   logits[b,o,x,y] = sum_{i,j} X[b,x,i] W1[i,o,j] X[b,y,j]
//                   + lin_i[b,x,o] + lin_j[b,y,o] + bias[o]
//   then row/col mask (-1e12) and strict lower-triangle (-1e12).
//
// One workgroup per (b, o, x-tile of 16):
//   phase 1: T[16][768] = X[b, x0:x0+16, :] * W1[:, o, :]  -> LDS (never hits HBM)
//   phase 2: for each y-tile: C = T * X[b]^T, fused epilogue, store to out.

typedef __attribute__((ext_vector_type(2))) float v2f;
typedef __attribute__((ext_vector_type(8))) float v8f;

#define B_   8
#define L_   512
#define H_   768
#define O_   12
#define OH_  (O_ * H_)        // 9216
#define LDT  772              // padded LDS row pitch (768+4): conflict-free ds_load_b64
#define NEGV 1000000000000.0f

static __device__ __forceinline__ v8f wmma4(v2f a, v2f b, v8f c) {
    // D = A(16x4,f32) * B(4x16,f32) + C(16x16,f32)
    return __builtin_amdgcn_wmma_f32_16x16x4_f32(
        /*neg_a=*/false, a, /*neg_b=*/false, b,
        /*c_mod=*/(short)0, c, /*reuse_a=*/false, /*reuse_b=*/false);
}

__launch_bounds__(256)
__global__ void biaffine_fused(const float* __restrict__ X,   // (B, L, H)
                               const float* __restrict__ W1,  // (H, O, H)
                               const float* __restrict__ W2,  // (2H+1, O)
                               const int*   __restrict__ M,   // (B, L)
                               float*       __restrict__ out) // (B, O, L, L)
{
    __shared__ float T[16 * LDT];   // stage-1 tile, 49,408 B
    __shared__ float LJ[L_];        // lin_j[b, y, o]
    __shared__ float MJf[L_];       // mask[b, y] as float
    __shared__ float LI[16];        // lin_i[b, x, o] + bias[o]
    __shared__ float MIf[16];       // mask[b, x] as float

    const int lane = threadIdx.x & 31;
    const int wave = threadIdx.x >> 5;   // 0..7
    const int half = lane >> 4;          // 0|1  -> K group / C row group
    const int l15  = lane & 15;          // A row / B,C column within tile

    const int wg = blockIdx.x;           // 0 .. B*O*32 - 1
    const int xt = wg & 31;              // x-tile index (L/16 == 32)
    const int bo = wg >> 5;
    const int o  = bo % O_;
    const int b  = bo / O_;
    const int x0 = xt * 16;

    const float* xb = X + (size_t)b * L_ * H_;

    // ---- pre-pass: linear terms + masks into LDS (pure VALU, tiny) ----
    for (int y = threadIdx.x; y < L_; y += 256) {
        const float* xr = xb + (size_t)y * H_;
        float s = 0.f;
        for (int i = 0; i < H_; ++i) s += xr[i] * W2[(H_ + i) * O_ + o];
        LJ[y]  = s;
        MJf[y] = (float)M[b * L_ + y];
    }
    if (threadIdx.x < 16) {
        const float* xr = xb + (size_t)(x0 + threadIdx.x) * H_;
        float s = W2[2 * H_ * O_ + o];               // fold bias into lin_i
        for (int i = 0; i < H_; ++i) s += xr[i] * W2[i * O_ + o];
        LI[threadIdx.x]  = s;
        MIf[threadIdx.x] = (float)M[b * L_ + x0 + threadIdx.x];
    }

    // ---- phase 1: T = X-rows (16 x H) * W1[:,o,:] (H x H), result to LDS ----
    // A layout (16x4 f32): lane<16 -> row l15, K=k..k+1 ; lane>=16 -> row l15, K=k+2..k+3
    const float* arow = xb + (size_t)(x0 + l15) * H_ + half * 2;
    for (int nt = wave; nt < H_ / 16; nt += 8) {       // 48 n-tiles, 6 per wave
        const int n0 = nt * 16;
        const float* bcol = W1 + (size_t)(half * 2) * OH_ + (size_t)o * H_ + n0 + l15;
        v8f acc0 = {}, acc1 = {};
        for (int k = 0; k < H_; k += 8) {              // 2 independent WMMA chains
            v2f a0 = *(const v2f*)(arow + k);
            v2f a1 = *(const v2f*)(arow + k + 4);
            const float* bp = bcol + (size_t)k * OH_;
            v2f b0, b1;
            b0.x = bp[0];                 b0.y = bp[OH_];
            b1.x = bp[4 * (size_t)OH_];   b1.y = bp[5 * (size_t)OH_];
            acc0 = wmma4(a0, b0, acc0);
            acc1 = wmma4(a1, b1, acc1);
        }
        // C layout: col = n0+l15, row = half*8 + v
        #pragma unroll
        for (int v = 0; v < 8; ++v)
            T[(half * 8 + v) * LDT + n0 + l15] = acc0[v] + acc1[v];
    }

    __syncthreads();

    // ---- phase 2: per y-tile, C(16x16) = T(16xH) * X[b]^T(Hx16) + epilogue ----
    const float* abase = &T[l15 * LDT + half * 2];     // LDS A reads: ds_load_b64
    for (int yt = wave; yt < L_ / 16; yt += 8) {       // 32 y-tiles, 4 per wave
        const int y0 = yt * 16;
        // B[k][n] = X[b, y0+n, k] : contiguous float2 in K, lanes stride H in n
        const float* bbase = xb + (size_t)(y0 + l15) * H_ + half * 2;
        v8f acc0 = {}, acc1 = {};
        for (int k = 0; k < H_; k += 8) {
            __builtin_prefetch(bbase + k + 64, 0, 0);  // global_prefetch_b8
            v2f a0 = *(const v2f*)(abase + k);
            v2f a1 = *(const v2f*)(abase + k + 4);
            v2f b0 = *(const v2f*)(bbase + k);
            v2f b1 = *(const v2f*)(bbase + k + 4);
            acc0 = wmma4(a0, b0, acc0);
            acc1 = wmma4(a1, b1, acc1);
        }
        const int   n  = y0 + l15;                     // global y
        const float lj = LJ[n];
        const float mc = MJf[n];
        float* orow = out + (((size_t)(b * O_ + o) * L_ + x0 + half * 8) * L_) + n;
        #pragma unroll
        for (int v = 0; v < 8; ++v) {
            const int   m   = half * 8 + v;            // local x row
            const float mr  = MIf[m];
            float val = acc0[v] + acc1[v] + LI[m] + lj;
            val = val * mr - NEGV * (1.f - mr);
            val = val * mc - NEGV * (1.f - mc);
            if (n < x0 + m) val -= NEGV;               // strict lower triangle
            orow[(size_t)v * L_] = val;
        }
    }
}

extern "C" void kernel_launch(void* const* d_in, const int* in_sizes, int n_in,
                              void* d_out, int out_size, void* d_ws, size_t ws_size,
                              hipStream_t stream) {
    const float* X  = (const float*)d_in[0];   // inputs  (8, 512, 768) f32
    const float* W1 = (const float*)d_in[1];   // weight1 (768, 12, 768) f32
    const float* W2 = (const float*)d_in[2];   // weight2 (1537, 12) f32
    const int*   M  = (const int*)d_in[3];     // mask    (8, 512) i32
    float* out = (float*)d_out;                // (8, 12, 512, 512) f32

    const int grid = B_ * O_ * (L_ / 16);      // 3072 workgroups
    biaffine_fused<<<grid, 256, 0, stream>>>(X, W1, W2, M, out);

    (void)in_sizes; (void)n_in; (void)out_size; (void)d_ws; (void)ws_size;
}